// Model_9242769621835
// MI455X (gfx1250) — compile-verified
//
#include <hip/hip_runtime.h>

typedef __attribute__((ext_vector_type(2))) float v2f;
typedef __attribute__((ext_vector_type(8))) float v8f;
typedef __attribute__((ext_vector_type(4))) unsigned uint4v;
typedef __attribute__((ext_vector_type(8))) int     int8v;
typedef __attribute__((ext_vector_type(4))) int     int4v;

#define B_  2
#define N_  8192
#define M_  8192
#define F_  128
#define ROWS_PER_BLOCK 128
#define THREADS 256
#define EPS_ 1e-6f
#define IDX_MASK 8191u

// ---------------------------------------------------------------------------
// Kernel 1: pack source points into [sx, sy, sz, |s|^2] float4
// ---------------------------------------------------------------------------
__global__ void pack_src_kernel(const float* __restrict__ src,
                                float4* __restrict__ packed, int total) {
    int i = blockIdx.x * blockDim.x + threadIdx.x;
    if (i < total) {
        float x = src[3 * i + 0];
        float y = src[3 * i + 1];
        float z = src[3 * i + 2];
        packed[i] = make_float4(x, y, z, x * x + y * y + z * z);
    }
}

// Monotonic f32 -> u32 total-order transform.
__device__ __forceinline__ unsigned f32_ord(float v) {
    int ib = __float_as_int(v);
    return (unsigned)ib ^ ((unsigned)(ib >> 31) | 0x80000000u);
}

// ---------------------------------------------------------------------------
// Kernel 2: WMMA 16x16x4_f32 brute-force 3-NN + inverse-distance interp.
//   key = (ord(dist_partial) & ~8191) | src_index  -> top-3 via min/max network
// ---------------------------------------------------------------------------
__global__ __launch_bounds__(THREADS)
void knn3_interp_kernel(const float4* __restrict__ spacked,
                        const float*  __restrict__ target_pc,
                        const float*  __restrict__ sfeat,
                        float*        __restrict__ out) {
    extern __shared__ char smem[];
    float4*   s_pts = (float4*)smem;                               // 128 KB @ 0
    unsigned* s_ck  = (unsigned*)(smem + (size_t)N_ * 16);         // 24 KB cand keys
    float*    s_w   = (float*)(smem + (size_t)N_ * 16 + 24576);    // 1.5 KB weights
    int*      s_i   = (int*)  (smem + (size_t)N_ * 16 + 26112);    // 1.5 KB indices

    const int tid  = threadIdx.x;
    const int lane = tid & 31;
    const int wave = tid >> 5;
    const int b    = blockIdx.x / (M_ / ROWS_PER_BLOCK);
    const int mblk = (blockIdx.x % (M_ / ROWS_PER_BLOCK)) * ROWS_PER_BLOCK;
    const int m0   = mblk + wave * 16;

    const float4* gsp = spacked + (size_t)b * N_;

#if __has_builtin(__builtin_amdgcn_tensor_load_to_lds) && __has_builtin(__builtin_amdgcn_s_wait_tensorcnt)
    // ---- Tensor Data Mover: DMA 128 KB of packed source points into LDS ----
    if (wave == 0) {
        unsigned long long ga = (unsigned long long)(const void*)gsp;
        // D# group 0: count=1 | lds_addr=0 | global_addr | type=2
        uint4v g0 = { 1u, 0u,
                      (unsigned)(ga & 0xFFFFFFFFu),
                      (unsigned)(ga >> 32) | (2u << 30) };
        // D# group 1: data_size=4B; tensor_dim0=32768, tensor_dim1=1;
        //             tile_dim0=32768, tile_dim1=1; dim0_stride=32768
        int8v g1 = { (int)0x00020000,        // data_size=2 (4B)
                     (int)0x80000000,        // tensor_dim0[15:0]=0x8000 in [31:16]
                     (int)0x00010000,        // tensor_dim1=1 in [31:16]
                     (int)0x80000000,        // tile_dim0=0x8000 in [31:16]
                     1,                      // tile_dim1=1
                     0x8000,                 // tensor_dim0_stride=32768
                     0, 0 };
        int4v g2 = { 0, 0, 0, 0 };
        int4v g3 = { 0, 0, 0, 0 };
        int8v g4 = { 0, 0, 0, 0, 0, 0, 0, 0 };
        __builtin_amdgcn_tensor_load_to_lds(g0, g1, g2, g3, g4, 0);
        __builtin_amdgcn_s_wait_tensorcnt(0);
    }
#else
    for (int i = tid; i < N_; i += THREADS) s_pts[i] = gsp[i];
#endif
    __syncthreads();

    // A fragment (16x4 f32, 2 VGPRs): A(m,:) = [-2tx, -2ty, -2tz, 1]
    const int mrow = m0 + (lane & 15);
    const float* tp = target_pc + ((size_t)b * M_ + mrow) * 3;
    v2f a;
    if (lane < 16) { a.x = -2.0f * tp[0]; a.y = -2.0f * tp[1]; }
    else           { a.x = -2.0f * tp[2]; a.y = 1.0f; }

    unsigned k0[8], k1[8], k2[8];
#pragma unroll
    for (int v = 0; v < 8; ++v) { k0[v] = k1[v] = k2[v] = 0xFFFFFFFFu; }

    // B fragment per tile: one ds_load_b64, software-pipelined one tile ahead.
    // (The prefetch at nb = N_-16 over-reads one tile into the candidate-key
    //  region -- in-bounds LDS, value never consumed.)
    const float2* s_p2 = (const float2*)s_pts;
    const int colbase = ((lane & 15) << 1) + (lane >> 4);

    float2 bb = s_p2[colbase];
#pragma unroll 2
    for (int nb = 0; nb < N_; nb += 16) {
        v2f bv; bv.x = bb.x; bv.y = bb.y;
        bb = s_p2[((nb + 16) << 1) + colbase];   // prefetch next tile

        v8f c = {};
        c = __builtin_amdgcn_wmma_f32_16x16x4_f32(
                false, a, false, bv, (short)0, c, false, false);

        const unsigned nidx = (unsigned)(nb + (lane & 15));
#pragma unroll
        for (int v = 0; v < 8; ++v) {
            unsigned key = (f32_ord(c[v]) & ~IDX_MASK) | nidx;
            unsigned lo = key < k0[v] ? key   : k0[v];
            unsigned hi = key < k0[v] ? k0[v] : key;
            k0[v] = lo;
            unsigned m1 = hi < k1[v] ? hi    : k1[v];
            unsigned h1 = hi < k1[v] ? k1[v] : hi;
            k1[v] = m1;
            k2[v] = h1 < k2[v] ? h1 : k2[v];
        }
    }

    // Dump per-lane stripe candidates: [wave][row 0..15][col 0..15][3] keys
    {
        const int col   = lane & 15;
        const int rbase = (lane >> 4) * 8;
#pragma unroll
        for (int v = 0; v < 8; ++v) {
            int o = ((wave * 16 + rbase + v) * 16 + col) * 3;
            s_ck[o + 0] = k0[v];
            s_ck[o + 1] = k1[v];
            s_ck[o + 2] = k2[v];
        }
    }
    __syncthreads();

    // Merge: lane r (<16) reduces 48 keys of its row to the final top-3, then
    // recomputes exact distances (t2 + s2 - 2 t.s) for the weights.
    if (lane < 16) {
        const int row = lane;
        const int m   = m0 + row;
        unsigned e0 = 0xFFFFFFFFu, e1 = 0xFFFFFFFFu, e2 = 0xFFFFFFFFu;
        const unsigned* ck = s_ck + (size_t)(wave * 16 + row) * 48;
        for (int t = 0; t < 48; ++t) {
            unsigned key = ck[t];
            unsigned lo = key < e0 ? key : e0;
            unsigned hi = key < e0 ? e0  : key;
            e0 = lo;
            unsigned m1 = hi < e1 ? hi : e1;
            unsigned h1 = hi < e1 ? e1 : hi;
            e1 = m1;
            e2 = h1 < e2 ? h1 : e2;
        }
        const int j0 = (int)(e0 & IDX_MASK);
        const int j1 = (int)(e1 & IDX_MASK);
        const int j2 = (int)(e2 & IDX_MASK);

        const float* tq = target_pc + ((size_t)b * M_ + m) * 3;
        float tx = tq[0], ty = tq[1], tz = tq[2];
        float t2 = tx * tx + ty * ty + tz * tz;
        float4 p0 = s_pts[j0];
        float4 p1 = s_pts[j1];
        float4 p2 = s_pts[j2];
        float q0 = t2 + p0.w - 2.0f * (tx * p0.x + ty * p0.y + tz * p0.z);
        float q1 = t2 + p1.w - 2.0f * (tx * p1.x + ty * p1.y + tz * p1.z);
        float q2 = t2 + p2.w - 2.0f * (tx * p2.x + ty * p2.y + tz * p2.z);
        float w0 = 1.0f / (sqrtf(fmaxf(q0, 0.0f)) + EPS_);
        float w1 = 1.0f / (sqrtf(fmaxf(q1, 0.0f)) + EPS_);
        float w2 = 1.0f / (sqrtf(fmaxf(q2, 0.0f)) + EPS_);
        float ws = w0 + w1 + w2;
        int lr = wave * 16 + row;
        s_w[lr * 3 + 0] = w0 / ws; s_w[lr * 3 + 1] = w1 / ws; s_w[lr * 3 + 2] = w2 / ws;
        s_i[lr * 3 + 0] = j0;      s_i[lr * 3 + 1] = j1;      s_i[lr * 3 + 2] = j2;
    }
    __syncthreads();

    // Gather: one wave per target row, 32 lanes x float4 = 128 feats, coalesced.
    const float4* feat4 = (const float4*)(sfeat + (size_t)b * N_ * F_);
    float4* out4 = (float4*)(out + ((size_t)b * M_ + mblk) * F_);
    for (int r = 0; r < 16; ++r) {
        const int lr = wave * 16 + r;
        const int j0 = s_i[lr * 3 + 0];
        const int j1 = s_i[lr * 3 + 1];
        const int j2 = s_i[lr * 3 + 2];
        const float w0 = s_w[lr * 3 + 0];
        const float w1 = s_w[lr * 3 + 1];
        const float w2 = s_w[lr * 3 + 2];
        float4 f0 = feat4[(size_t)j0 * (F_ / 4) + lane];
        float4 f1 = feat4[(size_t)j1 * (F_ / 4) + lane];
        float4 f2 = feat4[(size_t)j2 * (F_ / 4) + lane];
        float4 o;
        o.x = w0 * f0.x + w1 * f1.x + w2 * f2.x;
        o.y = w0 * f0.y + w1 * f1.y + w2 * f2.y;
        o.z = w0 * f0.z + w1 * f1.z + w2 * f2.z;
        o.w = w0 * f0.w + w1 * f1.w + w2 * f2.w;
        out4[(size_t)lr * (F_ / 4) + lane] = o;
    }
}

// ---------------------------------------------------------------------------
extern "C" void kernel_launch(void* const* d_in, const int* in_sizes, int n_in,
                              void* d_out, int out_size, void* d_ws, size_t ws_size,
                              hipStream_t stream) {
    (void)in_sizes; (void)n_in; (void)out_size; (void)ws_size;
    const float* src  = (const float*)d_in[0];   // [B,N,3]
    const float* tgt  = (const float*)d_in[1];   // [B,M,3]
    const float* feat = (const float*)d_in[2];   // [B,N,F]
    float* out = (float*)d_out;                  // [B,M,F]
    float4* packed = (float4*)d_ws;              // [B,N] float4 = 256 KB

    const int total = B_ * N_;
    pack_src_kernel<<<(total + 255) / 256, 256, 0, stream>>>(src, packed, total);

    const size_t shmem = (size_t)N_ * 16   // staged source points (128 KB)
                       + 24576             // candidate keys (24 KB)
                       + 2 * 1536;         // final weights + indices (3 KB)
    knn3_interp_kernel<<<B_ * (M_ / ROWS_PER_BLOCK), THREADS, shmem, stream>>>(
        packed, tgt, feat, out);
}